// VelocityHierarchicalGNN_34359738368488
// MI455X (gfx1250) — compile-verified
//
#include <hip/hip_runtime.h>

// ---------------------------------------------------------------------------
// VelocityHierarchicalGNN for MI455X (gfx1250, wave32, WMMA).
//
//  * density_mlp evaluated per NODE (32x less dense work on the fine graph),
//    gathered per edge through L2 (whole working set ~80MB < 192MB L2).
//  * All dense MLPs run through V_WMMA_F32_16X16X4_F32 with
//    REGISTER-RESIDENT weight fragments (templated shapes -> fully unrolled,
//    B fragments live in VGPRs; only the inter-layer activation transpose
//    goes through LDS).
//  * Final layer stays in WMMA C-registers; epilogue (bias / gather /
//    multiply / scatter-atomic or store) works directly on C layout.
//  * global_prefetch_b8 issued for the next grid-stride tile.
// ---------------------------------------------------------------------------

#define DPAD 32
#define WAVES_PER_BLOCK 8
#define BLOCK_THREADS (WAVES_PER_BLOCK * 32)

typedef __attribute__((ext_vector_type(2))) float v2f;
typedef __attribute__((ext_vector_type(8))) float v8f;

struct MlpPtrs {
    const float* W[3];
    const float* b[3];
    int dims[4];   // dims[0]=cin, dims[l+1]=cout of layer l
};

__device__ __forceinline__ v8f zero8() {
    v8f z = {0, 0, 0, 0, 0, 0, 0, 0};
    return z;
}

// Register-resident weight fragments for an NL-layer MLP.
// Layer 0 has K0 K-steps (K0 = ceil(cin/4)); later layers have 6 (cin=24).
template <int K0, int NL>
struct WReg {
    static constexpr int NSTEPS = K0 + (NL - 1) * 6;
    v2f w[NSTEPS][2];     // [kstep][ntile] B fragments
    float bias[NL][2];    // per-lane bias for each ntile
};

template <int K0, int NL>
__device__ __forceinline__ void load_weights(WReg<K0, NL>& wr, const MlpPtrs& p,
                                             int lane) {
    const int r = lane & 15;
    const int half = lane >> 4;
#pragma unroll
    for (int l = 0; l < NL; ++l) {
        const int cin = p.dims[l];
        const int cout = p.dims[l + 1];
        const float* W = p.W[l];
        const int kSteps = (l == 0) ? K0 : 6;
        const int base = (l == 0) ? 0 : K0 + (l - 1) * 6;
#pragma unroll
        for (int ks = 0; ks < kSteps; ++ks) {
            const int kb = (ks << 2) + (half << 1);
#pragma unroll
            for (int nt = 0; nt < 2; ++nt) {
                const int n = nt * 16 + r;
                v2f b;
                b.x = (kb < cin && n < cout) ? W[kb * cout + n] : 0.0f;
                b.y = (kb + 1 < cin && n < cout) ? W[(kb + 1) * cout + n] : 0.0f;
                wr.w[base + ks][nt] = b;
            }
        }
        wr.bias[l][0] = (r < cout) ? p.b[l][r] : 0.0f;
        wr.bias[l][1] = (16 + r < cout) ? p.b[l][16 + r] : 0.0f;
    }
}

// Run the whole MLP on one 16-row tile.  Input activations (layer0, padded
// K columns zeroed) are in act0; intermediate layers ping-pong through LDS
// for the C->A layout transpose; final layer result (bias added, no relu)
// is returned in c0/c1 in WMMA C layout.
template <int K0, int NL>
__device__ __forceinline__ void run_mlp_tile(float* act0, float* act1,
                                             const WReg<K0, NL>& wr, v8f& c0,
                                             v8f& c1, int lane) {
    const int r = lane & 15;
    const int half = lane >> 4;
    float* cur = act0;
    float* nxt = act1;
#pragma unroll
    for (int l = 0; l < NL; ++l) {
        const int kSteps = (l == 0) ? K0 : 6;
        const int base = (l == 0) ? 0 : K0 + (l - 1) * 6;
        c0 = zero8();
        c1 = zero8();
#pragma unroll
        for (int ks = 0; ks < kSteps; ++ks) {
            const int kb = (ks << 2) + (half << 1);
            v2f a;
            a.x = cur[r * DPAD + kb];
            a.y = cur[r * DPAD + kb + 1];
            c0 = __builtin_amdgcn_wmma_f32_16x16x4_f32(
                false, a, false, wr.w[base + ks][0], (short)0, c0, false, false);
            c1 = __builtin_amdgcn_wmma_f32_16x16x4_f32(
                false, a, false, wr.w[base + ks][1], (short)0, c1, false, false);
        }
        if (l < NL - 1) {
#pragma unroll
            for (int v = 0; v < 8; ++v) {
                const int m = v + (half << 3);
                nxt[m * DPAD + r] = fmaxf(c0[v] + wr.bias[l][0], 0.0f);
                nxt[m * DPAD + 16 + r] = fmaxf(c1[v] + wr.bias[l][1], 0.0f);
            }
            float* t = cur; cur = nxt; nxt = t;
        } else {
#pragma unroll
            for (int v = 0; v < 8; ++v) {
                c0[v] += wr.bias[l][0];
                c1[v] += wr.bias[l][1];
            }
        }
    }
}

// Generic row-wise MLP: Y[n, COUT] = mlp(X[n, cin]).
template <int K0, int NL, int COUT>
__global__ void node_mlp_kernel(const float* __restrict__ X,
                                float* __restrict__ Y, int n, MlpPtrs p) {
    __shared__ float sAct[WAVES_PER_BLOCK][2][16 * DPAD];
    const int lane = threadIdx.x & 31;
    const int wave = threadIdx.x >> 5;
    const int r = lane & 15;
    const int half = lane >> 4;

    WReg<K0, NL> wr;
    load_weights<K0, NL>(wr, p, lane);

    const int cin = p.dims[0];
    const int nTiles = (n + 15) >> 4;
    const int stride = gridDim.x * WAVES_PER_BLOCK;
    constexpr int KC = 4 * K0;  // live K columns of layer 0

    for (int t = blockIdx.x * WAVES_PER_BLOCK + wave; t < nTiles; t += stride) {
        float* act0 = sAct[wave][0];
        float* act1 = sAct[wave][1];
        const int row0 = t << 4;
        __builtin_prefetch(X + (size_t)(row0 + (stride << 4)) * cin, 0, 1);
        for (int i = lane; i < 16 * KC; i += 32) {
            const int rr = i / KC, cc = i % KC;
            const int row = row0 + rr;
            act0[rr * DPAD + cc] = (row < n && cc < cin) ? X[row * cin + cc] : 0.0f;
        }
        v8f c0, c1;
        run_mlp_tile<K0, NL>(act0, act1, wr, c0, c1, lane);
#pragma unroll
        for (int v = 0; v < 8; ++v) {
            const int row = row0 + v + (half << 3);
            if (row < n) {
                if (r < (COUT < 16 ? COUT : 16)) Y[row * COUT + r] = c0[v];
                if (COUT > 16 && r < COUT - 16) Y[row * COUT + 16 + r] = c1[v];
            }
        }
    }
}

// Fused per-edge: m = edge_mlp(edge_attr[e]) * density[src[e]];
// atomic-add into acc[dst[e]].  Edge MLP = 3 WMMA layers per 16-edge tile.
__global__ void edge_conv_kernel(const float* __restrict__ edge_attr,
                                 const int* __restrict__ src,
                                 const int* __restrict__ dst,
                                 const float* __restrict__ density,
                                 float* __restrict__ acc, int E, MlpPtrs p) {
    __shared__ float sAct[WAVES_PER_BLOCK][2][16 * DPAD];
    const int lane = threadIdx.x & 31;
    const int wave = threadIdx.x >> 5;
    const int r = lane & 15;
    const int half = lane >> 4;

    WReg<1, 3> wr;
    load_weights<1, 3>(wr, p, lane);

    const int nTiles = (E + 15) >> 4;
    const int stride = gridDim.x * WAVES_PER_BLOCK;
    for (int t = blockIdx.x * WAVES_PER_BLOCK + wave; t < nTiles; t += stride) {
        float* act0 = sAct[wave][0];
        float* act1 = sAct[wave][1];
        const int e0 = t << 4;
        __builtin_prefetch(edge_attr + (size_t)(e0 + (stride << 4)) * 3, 0, 1);
        // stage 16 edges x 4 cols (col 3 = zero pad)
        for (int i = lane; i < 64; i += 32) {
            const int rr = i >> 2, cc = i & 3;
            const int e = e0 + rr;
            act0[rr * DPAD + cc] = (e < E && cc < 3) ? edge_attr[e * 3 + cc] : 0.0f;
        }
        v8f c0, c1;
        run_mlp_tile<1, 3>(act0, act1, wr, c0, c1, lane);
        // gather density[src] * msg, scatter-atomic into acc[dst] from C regs
#pragma unroll
        for (int v = 0; v < 8; ++v) {
            const int e = e0 + v + (half << 3);
            if (e < E) {
                const int s = src[e];
                const int d = dst[e];
                atomicAdd(&acc[d * 24 + r], c0[v] * density[s * 24 + r]);
                if (r < 8)
                    atomicAdd(&acc[d * 24 + 16 + r],
                              c1[v] * density[s * 24 + 16 + r]);
            }
        }
    }
}

__global__ void degree_kernel(const int* __restrict__ dst,
                              float* __restrict__ deg, int E) {
    const int i = blockIdx.x * blockDim.x + threadIdx.x;
    if (i < E) atomicAdd(&deg[dst[i]], 1.0f);
}

__global__ void invdeg_kernel(float* deg, int n) {
    const int i = blockIdx.x * blockDim.x + threadIdx.x;
    if (i < n) deg[i] = 1.0f / fmaxf(deg[i], 1.0f);
}

__global__ void finalize_kernel(const float* __restrict__ acc,
                                const float* __restrict__ inv,
                                float* __restrict__ out, int n) {
    const int i = blockIdx.x * blockDim.x + threadIdx.x;
    if (i < n * 24) out[i] = acc[i] * inv[i / 24];
}

__global__ void combine_kernel(float* __restrict__ out,
                               const float* __restrict__ out_c,
                               const int* __restrict__ cluster, int N) {
    const int i = blockIdx.x * blockDim.x + threadIdx.x;
    if (i < N * 3) {
        const int node = i / 3, c = i % 3;
        out[i] += out_c[cluster[node] * 3 + c];
    }
}

static inline int imin_host(int a, int b) { return a < b ? a : b; }

// One GNN (3 EdgeConv layers + 2-layer readout) -> outPtr[n,3].
static void run_graph(const float* xin, const int* ei, const float* ea, int n,
                      int E, void* const* d_in, int convBase, int roBase,
                      float* outPtr, float* featA, float* featB, float* dens,
                      float* acc, float* inv, hipStream_t stream) {
    hipMemsetAsync(inv, 0, (size_t)n * sizeof(float), stream);
    degree_kernel<<<(E + 255) / 256, 256, 0, stream>>>(ei + E, inv, E);
    invdeg_kernel<<<(n + 255) / 256, 256, 0, stream>>>(inv, n);

    const float* cur = xin;
    float* bufs[2] = {featA, featB};
    int pb = 0;
    for (int cvi = 0; cvi < 3; ++cvi) {
        const int base = convBase + cvi * 12;
        MlpPtrs dp;  // density MLP: cin -> 24 -> 24 -> 24
        dp.W[0] = (const float*)d_in[base + 0];  dp.b[0] = (const float*)d_in[base + 1];
        dp.W[1] = (const float*)d_in[base + 2];  dp.b[1] = (const float*)d_in[base + 3];
        dp.W[2] = (const float*)d_in[base + 4];  dp.b[2] = (const float*)d_in[base + 5];
        dp.dims[0] = (cvi == 0) ? 3 : 24;
        dp.dims[1] = 24; dp.dims[2] = 24; dp.dims[3] = 24;

        MlpPtrs ep;  // edge MLP: 3 -> 24 -> 24 -> 24
        ep.W[0] = (const float*)d_in[base + 6];  ep.b[0] = (const float*)d_in[base + 7];
        ep.W[1] = (const float*)d_in[base + 8];  ep.b[1] = (const float*)d_in[base + 9];
        ep.W[2] = (const float*)d_in[base + 10]; ep.b[2] = (const float*)d_in[base + 11];
        ep.dims[0] = 3; ep.dims[1] = 24; ep.dims[2] = 24; ep.dims[3] = 24;

        const int nodeTiles = (n + 15) / 16;
        const int nodeBlocks =
            imin_host((nodeTiles + WAVES_PER_BLOCK - 1) / WAVES_PER_BLOCK, 2048);
        if (cvi == 0)
            node_mlp_kernel<1, 3, 24><<<nodeBlocks, BLOCK_THREADS, 0, stream>>>(
                cur, dens, n, dp);
        else
            node_mlp_kernel<6, 3, 24><<<nodeBlocks, BLOCK_THREADS, 0, stream>>>(
                cur, dens, n, dp);

        hipMemsetAsync(acc, 0, (size_t)n * 24 * sizeof(float), stream);
        const int edgeTiles = (E + 15) / 16;
        const int edgeBlocks =
            imin_host((edgeTiles + WAVES_PER_BLOCK - 1) / WAVES_PER_BLOCK, 4096);
        edge_conv_kernel<<<edgeBlocks, BLOCK_THREADS, 0, stream>>>(
            ea, ei, ei + E, dens, acc, E, ep);

        float* nextFeat = bufs[pb];
        pb ^= 1;
        finalize_kernel<<<(n * 24 + 255) / 256, 256, 0, stream>>>(acc, inv,
                                                                  nextFeat, n);
        cur = nextFeat;
    }

    MlpPtrs rp;  // readout: 24 -> 24 (relu) -> 3
    rp.W[0] = (const float*)d_in[roBase + 0]; rp.b[0] = (const float*)d_in[roBase + 1];
    rp.W[1] = (const float*)d_in[roBase + 2]; rp.b[1] = (const float*)d_in[roBase + 3];
    rp.W[2] = nullptr; rp.b[2] = nullptr;
    rp.dims[0] = 24; rp.dims[1] = 24; rp.dims[2] = 3; rp.dims[3] = 0;
    const int tiles = (n + 15) / 16;
    node_mlp_kernel<6, 2, 3>
        <<<imin_host((tiles + 7) / 8, 2048), BLOCK_THREADS, 0, stream>>>(
            cur, outPtr, n, rp);
}

extern "C" void kernel_launch(void* const* d_in, const int* in_sizes, int n_in,
                              void* d_out, int out_size, void* d_ws,
                              size_t ws_size, hipStream_t stream) {
    const int N = in_sizes[0] / 3;
    const int E = in_sizes[1] / 2;
    const int Nc = in_sizes[3] / 3;
    const int Ec = in_sizes[4] / 2;

    const float* x = (const float*)d_in[0];
    const int* ei = (const int*)d_in[1];
    const float* ea = (const float*)d_in[2];
    const float* xc = (const float*)d_in[3];
    const int* eic = (const int*)d_in[4];
    const float* eac = (const float*)d_in[5];
    const int* cluster = (const int*)d_in[6];

    // pytree leaf order: fine_convs (36), coarse_convs (36), readout_fine (4),
    // readout_coarse (4); dict keys sort "density" < "edge".
    const int fineBase = 7, coarseBase = 43, roFine = 79, roCoarse = 83;

    float* out = (float*)d_out;             // [N,3] then [Nc,3] flattened
    float* outc = out + (size_t)N * 3;

    char* ws = (char*)d_ws;
    size_t off = 0;
    auto carve = [&](size_t bytes) -> float* {
        float* p = (float*)(ws + off);
        off += (bytes + 255) & ~(size_t)255;
        return p;
    };
    float* featA = carve((size_t)N * 24 * sizeof(float));
    float* featB = carve((size_t)N * 24 * sizeof(float));
    float* dens = carve((size_t)N * 24 * sizeof(float));
    float* acc = carve((size_t)N * 24 * sizeof(float));
    float* inv = carve((size_t)N * sizeof(float));
    (void)ws_size; (void)n_in; (void)out_size;

    // Coarse GNN -> out_c
    run_graph(xc, eic, eac, Nc, Ec, d_in, coarseBase, roCoarse, outc, featA,
              featB, dens, acc, inv, stream);
    // Fine GNN -> correction term (written straight into out)
    run_graph(x, ei, ea, N, E, d_in, fineBase, roFine, out, featA, featB, dens,
              acc, inv, stream);
    // out += out_c[cluster]
    combine_kernel<<<(N * 3 + 255) / 256, 256, 0, stream>>>(out, outc, cluster, N);
}